// CompositeLoss_26173530702335
// MI455X (gfx1250) — compile-verified
//
#include <hip/hip_runtime.h>

typedef __attribute__((ext_vector_type(2))) float v2f;
typedef __attribute__((ext_vector_type(8))) float v8f;

#define IMG   512
#define OUTD  502          // 512 - 11 + 1 (VALID)
#define HALO  26           // 16 + 10
#define ROWS  32           // padded rows (two 16-row WMMA blocks)
#define COLS  28           // padded cols (7 K-steps of 4)
#define TSTRIDE 17         // sT row stride (bank-conflict pad)
#define TPLANE  (ROWS * TSTRIDE)
#define TILES_1D 32        // ceil(502/16)
#define TILES_PER_IMG (TILES_1D * TILES_1D)
#define MSE_BLOCKS 1024

__device__ __forceinline__ float clip01(float v) {
    return __builtin_fminf(__builtin_fmaxf(v, 0.f), 1.f);
}

// One wave (32 threads) computes one 16x16 SSIM output tile via WMMA GEMMs.
//   horizontal: T_m(26x16) = M_m(26x26) x Gband(26x16),  Gband[k][n] = g[k-n]
//   vertical:   O_m(16x16) = Gband^T(16x26) x T_m(26x16)
// Both band operands reduce to the same per-lane value g[k - (lane&15)]
// under the 16x16x4 fragment layouts, so one register set serves both.
__global__ __launch_bounds__(32)
void ssim_tile_kernel(const float* __restrict__ pred,
                      const float* __restrict__ targ,
                      float* __restrict__ ssim_partials)
{
    __shared__ float sX[ROWS * COLS];     // zero-padded clipped pred tile
    __shared__ float sY[ROWS * COLS];     // zero-padded clipped target tile
    __shared__ float sT[5 * TPLANE];      // horizontal-pass results, 5 maps
    __shared__ float sG[12];              // gaussian weights + trailing 0
    __shared__ float sRed[32];

    const int lane  = threadIdx.x;
    const int lid   = lane & 15;          // N index (B/C/D) or M index (A)
    const int lhalf = lane >> 4;          // selects K half

    const int bid = blockIdx.x;
    const int img = bid >> 10;            // 1024 tiles per image
    const int t   = bid & 1023;
    const int h0  = (t >> 5) * 16;
    const int w0  = (t & 31) * 16;

    if (lane < 12) {
        const float g[12] = {
            0.00102838f, 0.00759867f, 0.03600080f, 0.10936070f,
            0.21300578f, 0.26601198f, 0.21300578f, 0.10936070f,
            0.03600080f, 0.00759867f, 0.00102838f, 0.0f };
        sG[lane] = g[lane];
    }

    // zero-fill padded tiles, then load the 26x26 interior (clipped)
    for (int idx = lane; idx < ROWS * COLS; idx += 32) {
        sX[idx] = 0.f;
        sY[idx] = 0.f;
    }
    __syncthreads();

    const float* pX = pred + (size_t)img * (IMG * IMG);
    const float* pY = targ + (size_t)img * (IMG * IMG);
    for (int idx = lane; idx < HALO * HALO; idx += 32) {
        int r = idx / HALO, c = idx - r * HALO;
        int gh = h0 + r, gw = w0 + c;
        if (gh < IMG && gw < IMG) {
            size_t o = (size_t)gh * IMG + gw;
            sX[r * COLS + c] = clip01(pX[o]);
            sY[r * COLS + c] = clip01(pY[o]);
        }
    }
    __syncthreads();

    // ---- hoist the 7 band fragments into registers (reused by all WMMAs)
    v2f bw[7];
    #pragma unroll
    for (int kk = 0; kk < 7; ++kk) {
        #pragma unroll
        for (int v = 0; v < 2; ++v) {
            const int k  = kk * 4 + lhalf * 2 + v;
            const int d  = k - lid;
            const int dc = ((unsigned)d > 10u) ? 11 : d;   // sG[11] == 0
            if (v == 0) bw[kk].x = sG[dc];
            else        bw[kk].y = sG[dc];
        }
    }

    // ---- horizontal pass: per row-block, load X/Y fragments once,
    //      derive all 5 map operands with VALU, 7 WMMAs per map.
    #pragma unroll
    for (int rb = 0; rb < 2; ++rb) {
        const int rowbase = (rb * 16 + lid) * COLS;   // A: M index = lid
        v2f xa[7], ya[7];
        #pragma unroll
        for (int kk = 0; kk < 7; ++kk) {
            const int k0 = kk * 4 + lhalf * 2;
            xa[kk].x = sX[rowbase + k0];
            xa[kk].y = sX[rowbase + k0 + 1];
            ya[kk].x = sY[rowbase + k0];
            ya[kk].y = sY[rowbase + k0 + 1];
        }
        #pragma unroll
        for (int m = 0; m < 5; ++m) {
            v8f acc = {};
            #pragma unroll
            for (int kk = 0; kk < 7; ++kk) {
                v2f a;
                if      (m == 0) a = xa[kk];
                else if (m == 1) a = ya[kk];
                else if (m == 2) a = xa[kk] * xa[kk];
                else if (m == 3) a = ya[kk] * ya[kk];
                else             a = xa[kk] * ya[kk];
                acc = __builtin_amdgcn_wmma_f32_16x16x4_f32(
                          false, a, false, bw[kk], (short)0, acc, false, false);
            }
            #pragma unroll
            for (int r = 0; r < 8; ++r)
                sT[m * TPLANE + (rb * 16 + r + lhalf * 8) * TSTRIDE + lid] = acc[r];
        }
    }
    __syncthreads();

    // ---- vertical pass: band fragments as A, T planes as B
    v8f res[5];
    #pragma unroll
    for (int m = 0; m < 5; ++m) {
        v8f acc = {};
        #pragma unroll
        for (int kk = 0; kk < 7; ++kk) {
            const int k0 = kk * 4 + lhalf * 2;
            v2f b;
            b.x = sT[m * TPLANE + k0 * TSTRIDE + lid];       // rows >=26 are 0
            b.y = sT[m * TPLANE + (k0 + 1) * TSTRIDE + lid];
            acc = __builtin_amdgcn_wmma_f32_16x16x4_f32(
                      false, bw[kk], false, b, (short)0, acc, false, false);
        }
        res[m] = acc;
    }

    // ---- per-pixel SSIM map + masked tile sum
    const float C1 = 1.0e-4f;   // (0.01*1.0)^2
    const float C2 = 9.0e-4f;   // (0.03*1.0)^2
    float sum = 0.f;
    #pragma unroll
    for (int r = 0; r < 8; ++r) {
        const int row = r + lhalf * 8;                       // D layout
        const int h = h0 + row, w = w0 + lid;
        const float mu1 = res[0][r], mu2 = res[1][r];
        const float ex2 = res[2][r], ey2 = res[3][r], exy = res[4][r];
        const float mu1s = mu1 * mu1, mu2s = mu2 * mu2, mu12 = mu1 * mu2;
        const float s1  = ex2 - mu1s;
        const float s2  = ey2 - mu2s;
        const float s12 = exy - mu12;
        const float cs   = (2.f * s12 + C2) / (s1 + s2 + C2);
        const float ssim = ((2.f * mu12 + C1) / (mu1s + mu2s + C1)) * cs;
        if (h < OUTD && w < OUTD) sum += ssim;
    }

    sRed[lane] = sum;
    __syncthreads();
    if (lane == 0) {
        float s = 0.f;
        #pragma unroll
        for (int i = 0; i < 32; ++i) s += sRed[i];
        ssim_partials[bid] = s;
    }
}

__global__ __launch_bounds__(256)
void mse_kernel(const float* __restrict__ p, const float* __restrict__ t,
                float* __restrict__ partials, int n)
{
    __shared__ float red[256];
    float s = 0.f;
    for (int i = blockIdx.x * 256 + threadIdx.x; i < n; i += gridDim.x * 256) {
        float d = clip01(p[i]) - clip01(t[i]);
        s += d * d;
    }
    red[threadIdx.x] = s;
    __syncthreads();
    for (int o = 128; o > 0; o >>= 1) {
        if (threadIdx.x < o) red[threadIdx.x] += red[threadIdx.x + o];
        __syncthreads();
    }
    if (threadIdx.x == 0) partials[blockIdx.x] = red[0];
}

__global__ __launch_bounds__(256)
void final_kernel(const float* __restrict__ ssimP, int nS,
                  const float* __restrict__ mseP, int nM,
                  float invM, float invN, float* __restrict__ out)
{
    __shared__ float rs[256], rm[256];
    float ss = 0.f, ms = 0.f;
    for (int i = threadIdx.x; i < nS; i += 256) ss += ssimP[i];
    for (int i = threadIdx.x; i < nM; i += 256) ms += mseP[i];
    rs[threadIdx.x] = ss;
    rm[threadIdx.x] = ms;
    __syncthreads();
    for (int o = 128; o > 0; o >>= 1) {
        if (threadIdx.x < o) {
            rs[threadIdx.x] += rs[threadIdx.x + o];
            rm[threadIdx.x] += rm[threadIdx.x + o];
        }
        __syncthreads();
    }
    if (threadIdx.x == 0) {
        const float alpha = 0.84f;
        float mse   = rm[0] * invN;
        float mssim = rs[0] * invM;
        out[0] = (1.f - alpha) * mse + alpha * (1.f - mssim);
    }
}

extern "C" void kernel_launch(void* const* d_in, const int* in_sizes, int n_in,
                              void* d_out, int out_size, void* d_ws, size_t ws_size,
                              hipStream_t stream)
{
    (void)n_in; (void)out_size; (void)ws_size;
    const float* pred = (const float*)d_in[0];
    const float* targ = (const float*)d_in[1];
    float* out = (float*)d_out;

    const int n    = in_sizes[0];                 // 16*3*512*512
    const int nImg = n / (IMG * IMG);             // 48
    const int nS   = nImg * TILES_PER_IMG;        // 49152 ssim tiles

    float* ws    = (float*)d_ws;
    float* ssimP = ws;                            // [nS]
    float* mseP  = ws + nS;                       // [MSE_BLOCKS]

    ssim_tile_kernel<<<nS, 32, 0, stream>>>(pred, targ, ssimP);
    mse_kernel<<<MSE_BLOCKS, 256, 0, stream>>>(pred, targ, mseP, n);

    const float invM = 1.f / ((float)nImg * (float)OUTD * (float)OUTD);
    const float invN = 1.f / (float)n;
    final_kernel<<<1, 256, 0, stream>>>(ssimP, nS, mseP, MSE_BLOCKS, invM, invN, out);
}